// KinematicLayer_34239479284008
// MI455X (gfx1250) — compile-verified
//
#include <hip/hip_runtime.h>
#include <stdint.h>

#define TPB 256
#define JN 21
#define FPS 63                       // floats per sample (21 joints * 3)
#define BLK_FLOATS (TPB * FPS)       // 16128 floats per block per array
#define BLK_V4 (BLK_FLOATS / 4)      // 4032 16-byte packets
#define EPSF 1e-8f

// ---- CDNA5 async global<->LDS (ASYNCcnt) ------------------------------------
__device__ __forceinline__ void async_load16(uint32_t lds_byte_off, uint64_t gaddr) {
  asm volatile("global_load_async_to_lds_b128 %0, %1, off"
               :: "v"(lds_byte_off), "v"(gaddr) : "memory");
}
__device__ __forceinline__ void async_store16(uint64_t gaddr, uint32_t lds_byte_off) {
  asm volatile("global_store_async_from_lds_b128 %0, %1, off"
               :: "v"(gaddr), "v"(lds_byte_off) : "memory");
}
__device__ __forceinline__ void wait_async0() {
  asm volatile("s_wait_asynccnt 0" ::: "memory");
}

__global__ __launch_bounds__(TPB, 1)
void mano_fk_kernel(const float* __restrict__ joints,
                    const float* __restrict__ tempJ,
                    float* __restrict__ out, int n) {
  __shared__ float sJ[BLK_FLOATS];   // joints staging, later reused for output
  __shared__ float sT[BLK_FLOATS];   // template staging

  const int tid = threadIdx.x;
  const int base_sample = blockIdx.x * TPB;
  int nsamp = n - base_sample; if (nsamp > TPB) nsamp = TPB;
  if (nsamp <= 0) return;
  const int nflt = nsamp * FPS;
  const int nv4  = nflt >> 2;

  const uint64_t gJ = (uint64_t)(uintptr_t)joints + (uint64_t)base_sample * FPS * 4u;
  const uint64_t gT = (uint64_t)(uintptr_t)tempJ  + (uint64_t)base_sample * FPS * 4u;
  const uint32_t lJ = (uint32_t)(uintptr_t)(&sJ[0]);
  const uint32_t lT = (uint32_t)(uintptr_t)(&sT[0]);

  // Coalesced async staging: 16B per lane per op, straight into LDS.
  for (int i = tid; i < nv4; i += TPB) {
    async_load16(lJ + (uint32_t)i * 16u, gJ + (uint64_t)i * 16u);
    async_load16(lT + (uint32_t)i * 16u, gT + (uint64_t)i * 16u);
  }
  // scalar tail (only for a partial last block)
  for (int i = (nv4 << 2) + tid; i < nflt; i += TPB) {
    sJ[i] = joints[(size_t)base_sample * FPS + i];
    sT[i] = tempJ [(size_t)base_sample * FPS + i];
  }
  wait_async0();
  __syncthreads();

  if (tid < nsamp) {
    const float* Jb = &sJ[tid * FPS];
    const float* Tb = &sT[tid * FPS];

    const float wx = Jb[0], wy = Jb[1], wz = Jb[2];
    const float t0x = Tb[0], t0y = Tb[1], t0z = Tb[2];

    // Palm correlation H = sum_k src_k dst_k^T, src = template-rel, dst = wrist-rel.
    // Joint 0 contributes zero (both relative to index 0), so use {1,4,7,10,13}.
    float S00=0,S01=0,S02=0,S10=0,S11=0,S12=0,S20=0,S21=0,S22=0;
    {
      constexpr int PALM5[5] = {1,4,7,10,13};
      #pragma unroll
      for (int p = 0; p < 5; ++p) {
        const int k = PALM5[p] * 3;
        const float sx=Tb[k]-t0x, sy=Tb[k+1]-t0y, sz=Tb[k+2]-t0z;
        const float dx=Jb[k]-wx,  dy=Jb[k+1]-wy,  dz=Jb[k+2]-wz;
        S00+=sx*dx; S01+=sx*dy; S02+=sx*dz;
        S10+=sy*dx; S11+=sy*dy; S12+=sy*dz;
        S20+=sz*dx; S21+=sz*dy; S22+=sz*dz;
      }
    }
    // Horn 4x4 (max eigenvector = optimal proper rotation == Kabsch w/ det fix)
    const float N00=S00+S11+S22;
    const float N01=S12-S21, N02=S20-S02, N03=S01-S10;
    const float N11=S00-S11-S22, N22=S11-S00-S22, N33=S22-S00-S11;
    const float N12=S01+S10, N13=S02+S20, N23=S12+S21;
    const float fro = N00*N00+N11*N11+N22*N22+N33*N33
                    + 2.f*(N01*N01+N02*N02+N03*N03+N12*N12+N13*N13+N23*N23);
    const float sig = sqrtf(fro) + 1e-6f;   // shift -> PSD, dominant top eigenpair
    const float A00=N00+sig, A11=N11+sig, A22=N22+sig, A33=N33+sig;

    float q0=1.f, q1=0.0173f, q2=0.0137f, q3=0.0097f;   // avoid orthogonal start
    #pragma unroll
    for (int it = 0; it < 24; ++it) {
      float r0=A00*q0+N01*q1+N02*q2+N03*q3;
      float r1=N01*q0+A11*q1+N12*q2+N13*q3;
      float r2=N02*q0+N12*q1+A22*q2+N23*q3;
      float r3=N03*q0+N13*q1+N23*q2+A33*q3;
      if ((it & 3) == 3) {
        const float inv = rsqrtf(r0*r0+r1*r1+r2*r2+r3*r3+1e-30f);
        r0*=inv; r1*=inv; r2*=inv; r3*=inv;
      }
      q0=r0; q1=r1; q2=r2; q3=r3;
    }
    {
      const float inv = rsqrtf(q0*q0+q1*q1+q2*q2+q3*q3+1e-30f);
      q0*=inv; q1*=inv; q2*=inv; q3*=inv;
    }
    const float R00=1.f-2.f*(q2*q2+q3*q3), R01=2.f*(q1*q2-q0*q3), R02=2.f*(q1*q3+q0*q2);
    const float R10=2.f*(q1*q2+q0*q3), R11=1.f-2.f*(q1*q1+q3*q3), R12=2.f*(q2*q3-q0*q1);
    const float R20=2.f*(q1*q3-q0*q2), R21=2.f*(q2*q3+q0*q1), R22=1.f-2.f*(q1*q1+q2*q2);

    // T = R @ (template - template0): lives entirely in VGPRs (fully unrolled).
    float T[JN][3];
    #pragma unroll
    for (int k = 0; k < JN; ++k) {
      const float sx=Tb[k*3]-t0x, sy=Tb[k*3+1]-t0y, sz=Tb[k*3+2]-t0z;
      T[k][0]=R00*sx+R01*sy+R02*sz;
      T[k][1]=R10*sx+R11*sy+R12*sz;
      T[k][2]=R20*sx+R21*sy+R22*sz;
    }

    constexpr int CH[15]={2,3,17,5,6,18,8,9,20,11,12,19,14,15,16};
    constexpr int PA[15]={1,2,3,4,5,6,7,8,9,10,11,12,13,14,15};
    constexpr int FG[5][5]={{0,1,2,3,17},{0,4,5,6,18},{0,7,8,9,20},
                            {0,10,11,12,19},{0,13,14,15,16}};
    #pragma unroll
    for (int idx = 0; idx < 15; ++idx) {
      const int ch = CH[idx], pa = PA[idx];
      // --- bone length clamp ---
      const float jx=Jb[ch*3]-wx, jy=Jb[ch*3+1]-wy, jz=Jb[ch*3+2]-wz;
      const float v1x=jx-T[pa][0], v1y=jy-T[pa][1], v1z=jz-T[pa][2];
      const float dis = sqrtf(v1x*v1x+v1y*v1y+v1z*v1z);
      const float inv = 1.f/(dis+EPSF);
      const float bx=Tb[ch*3]-Tb[pa*3], by=Tb[ch*3+1]-Tb[pa*3+1], bz=Tb[ch*3+2]-Tb[pa*3+2];
      const float blen = sqrtf(bx*bx+by*by+bz*bz);
      const bool valid = (blen <= dis) && (dis <= blen);
      const float sc = blen*inv;
      T[ch][0] = valid ? jx : T[pa][0]+v1x*sc;
      T[ch][1] = valid ? jy : T[pa][1]+v1y*sc;
      T[ch][2] = valid ? jz : T[pa][2]+v1z*sc;
      // --- flex clamp (trig-free exact form) ---
      const int gp=FG[idx/3][idx%3], pp=FG[idx/3][idx%3+1], cc=FG[idx/3][idx%3+2];
      const float v0x=T[pp][0]-T[gp][0], v0y=T[pp][1]-T[gp][1], v0z=T[pp][2]-T[gp][2];
      const float vbx=T[cc][0]-T[pp][0], vby=T[cc][1]-T[pp][1], vbz=T[cc][2]-T[pp][2];
      const float crx=v0y*vbz-v0z*vby, cry=v0z*vbx-v0x*vbz, crz=v0x*vby-v0y*vbx;
      const float s2 = crx*crx+cry*cry+crz*crz;
      const float sn = sqrtf(s2);
      const float c  = v0x*vbx+v0y*vby+v0z*vbz;
      // ang = atan2(sn,c), sn>=0.  delta = min(ang,pi/2)-ang nonzero iff c<0;
      // then cos(delta)=sn/r, sin(delta)=c/r with r=sqrt(sn^2+c^2).
      const float rin = 1.f/sqrtf(s2 + c*c);      // inf only when selected away
      const bool flex = (c < 0.f);
      const float cd = flex ? sn*rin : 1.f;
      const float sd = flex ? c*rin  : 0.f;
      const float ain = 1.f/(sn+EPSF);
      const float ax=crx*ain, ay=cry*ain, az=crz*ain;
      const float adv = ax*vbx+ay*vby+az*vbz;
      const float k1  = adv*(1.f-cd);
      const float cx=ay*vbz-az*vby, cy=az*vbx-ax*vbz, cz=ax*vby-ay*vbx;
      T[cc][0]=vbx*cd+cx*sd+ax*k1+T[pp][0];
      T[cc][1]=vby*cd+cy*sd+ay*k1+T[pp][1];
      T[cc][2]=vbz*cd+cz*sd+az*k1+T[pp][2];
    }

    // Write results into my own sJ slice (only this lane touches it).
    #pragma unroll
    for (int k = 0; k < JN; ++k) {
      sJ[tid*FPS + k*3    ] = wx + T[k][0];
      sJ[tid*FPS + k*3 + 1] = wy + T[k][1];
      sJ[tid*FPS + k*3 + 2] = wz + T[k][2];
    }
  }
  __syncthreads();

  // Coalesced async drain LDS -> global.
  const uint64_t gO = (uint64_t)(uintptr_t)out + (uint64_t)base_sample * FPS * 4u;
  for (int i = tid; i < nv4; i += TPB) {
    async_store16(gO + (uint64_t)i * 16u, lJ + (uint32_t)i * 16u);
  }
  for (int i = (nv4 << 2) + tid; i < nflt; i += TPB) {
    out[(size_t)base_sample * FPS + i] = sJ[i];
  }
  wait_async0();
}

extern "C" void kernel_launch(void* const* d_in, const int* in_sizes, int n_in,
                              void* d_out, int out_size, void* d_ws, size_t ws_size,
                              hipStream_t stream) {
  (void)n_in; (void)out_size; (void)d_ws; (void)ws_size;
  const float* joints = (const float*)d_in[0];
  const float* tempJ  = (const float*)d_in[1];
  float* out = (float*)d_out;
  const int n = in_sizes[0] / FPS;           // samples
  if (n <= 0) return;
  const int blocks = (n + TPB - 1) / TPB;
  hipLaunchKernelGGL(mano_fk_kernel, dim3(blocks), dim3(TPB), 0, stream,
                     joints, tempJ, out, n);
}